// SpaceLoss2_88656714924204
// MI455X (gfx1250) — compile-verified
//
#include <hip/hip_runtime.h>
#include <math.h>

typedef __attribute__((ext_vector_type(2))) float v2f;
typedef __attribute__((ext_vector_type(8))) float v8f;
typedef __attribute__((ext_vector_type(4))) int   v4i;
typedef v4i __attribute__((address_space(3)))     lds_v4i;

#define BN       4096
#define DK       256
#define ROWS_PB  32
#define NTHREADS 64
#define LDA      260            // padded LDS row stride (floats): conflict-free, 16B-aligned rows
#define NCTILES  (BN / 16)
#define KSTEPS   (DK / 4)

// LDS layout (dynamic): As[32][LDA] | Bs[2][16][LDA] | negSh[32]
#define SMEM_FLOATS (ROWS_PB * LDA + 2 * 16 * LDA + 32)

#if defined(__AMDGCN__) &&                                                   \
    __has_builtin(__builtin_amdgcn_global_load_async_to_lds_b128) &&         \
    __has_builtin(__builtin_amdgcn_s_wait_asynccnt)
#define USE_ASYNC_LDS 1
#else
#define USE_ASYNC_LDS 0
#endif

// Load one 16-row x 256-col f32 tile of ftr into LDS (16B per lane per issue).
__device__ __forceinline__ void load_tile16(const float* src, float* dstBase, int tid)
{
    for (int i = tid; i < 16 * (DK / 4); i += NTHREADS) {
        const int r  = i >> 6;
        const int k4 = i & 63;
#if USE_ASYNC_LDS
        __builtin_amdgcn_global_load_async_to_lds_b128(
            (v4i*)(src + (size_t)r * DK + k4 * 4),
            (lds_v4i*)(v4i*)(dstBase + r * LDA + k4 * 4),
            /*offset=*/0, /*cpol=*/0);
#else
        const float4 v = reinterpret_cast<const float4*>(src + (size_t)r * DK)[k4];
        *reinterpret_cast<float4*>(dstBase + r * LDA + k4 * 4) = v;
#endif
    }
}

// Fused: Gram-matrix tiles via v_wmma_f32_16x16x4_f32 with double-buffered
// (async) B tiles, branchless running masked argmin, then per-row loss.
__global__ __launch_bounds__(NTHREADS)
void SpaceLoss2_wmma_kernel(const float* __restrict__ ftr,
                            const float* __restrict__ prototypes,
                            const int*   __restrict__ label,
                            float*       __restrict__ losses)
{
    extern __shared__ float smem[];
    float* As    = smem;                              // [ROWS_PB][LDA]
    float* Bs    = smem + ROWS_PB * LDA;              // [2][16][LDA]
    int*   negSh = (int*)(smem + ROWS_PB * LDA + 2 * 16 * LDA);  // [ROWS_PB]

    const int tid  = threadIdx.x;
    const int lane = tid & 31;
    const int wave = tid >> 5;
    const int l15  = lane & 15;
    const int half = lane >> 4;      // WMMA f32 A/B layout: half 0 -> K={0,1}, half 1 -> K={2,3}
    const int koff = half * 2;
    const int rowBlock = blockIdx.x * ROWS_PB;

    // Stage this block's 32 A rows into LDS (coalesced float4).
    for (int i = tid; i < ROWS_PB * (DK / 4); i += NTHREADS) {
        const int r  = i >> 6;
        const int k4 = i & 63;
        const float4 v =
            reinterpret_cast<const float4*>(ftr + (size_t)(rowBlock + r) * DK)[k4];
        *reinterpret_cast<float4*>(As + r * LDA + k4 * 4) = v;
    }

    // Labels of the 8 sim-rows this lane's accumulator covers
    // (C/D layout: VGPR r, lane l -> M = r + 8*(l>=16), N = l%16).
    int myLabel[8];
#pragma unroll
    for (int r = 0; r < 8; ++r)
        myLabel[r] = label[rowBlock + wave * 16 + half * 8 + r];

    float bestV[8];
    int   bestI[8];
#pragma unroll
    for (int r = 0; r < 8; ++r) { bestV[r] = INFINITY; bestI[r] = 0x7fffffff; }

    const float* aPtr = As + (size_t)(wave * 16 + l15) * LDA;  // A: M = lane%16

    // Prologue: tile 0 into buffer 0.
    load_tile16(ftr, Bs, tid);
#if USE_ASYNC_LDS
    __builtin_amdgcn_s_wait_asynccnt(0);
#endif
    __syncthreads();

    for (int ct = 0; ct < NCTILES; ++ct) {
        const int buf = ct & 1;
        // Prefetch next tile into the other buffer while we compute on this one.
        if (ct + 1 < NCTILES)
            load_tile16(ftr + (size_t)(ct + 1) * 16 * DK, Bs + (buf ^ 1) * 16 * LDA, tid);

        const float* bPtr = Bs + (size_t)buf * 16 * LDA + (size_t)l15 * LDA;

        // 16x16 sim tile = A(16xK) * B(Kx16), K=256 in 64 WMMA f32 16x16x4 steps.
        v8f acc = {0.f, 0.f, 0.f, 0.f, 0.f, 0.f, 0.f, 0.f};
#pragma unroll 8
        for (int ks = 0; ks < KSTEPS; ++ks) {
            v2f a = *reinterpret_cast<const v2f*>(aPtr + ks * 4 + koff);
            v2f b = *reinterpret_cast<const v2f*>(bPtr + ks * 4 + koff);
            acc = __builtin_amdgcn_wmma_f32_16x16x4_f32(
                /*neg_a=*/false, a, /*neg_b=*/false, b,
                /*c_mod=*/(short)0, acc, /*reuse_a=*/false, /*reuse_b=*/false);
        }

        // Branchless running masked argmin (mask: batch index == label[row];
        // first occurrence wins ties).
        const int col = ct * 16 + l15;
#pragma unroll
        for (int r = 0; r < 8; ++r) {
            const float v = acc[r];
            const bool take = (col != myLabel[r]) &
                              ((v < bestV[r]) | ((v == bestV[r]) & (col < bestI[r])));
            bestV[r] = take ? v : bestV[r];
            bestI[r] = take ? col : bestI[r];
        }

#if USE_ASYNC_LDS
        __builtin_amdgcn_s_wait_asynccnt(0);   // next tile landed in LDS
#endif
        __syncthreads();
    }

    // Cross-lane argmin reduction: each half-wave (16 lanes) holds the same 8
    // rows over disjoint column classes -> width-16 xor-shuffle tree.
#pragma unroll
    for (int r = 0; r < 8; ++r) {
        float bv = bestV[r];
        int   bi = bestI[r];
#pragma unroll
        for (int off = 8; off > 0; off >>= 1) {
            const float ov = __shfl_xor(bv, off, 16);
            const int   oi = __shfl_xor(bi, off, 16);
            const bool take = (ov < bv) | ((ov == bv) & (oi < bi));
            bv = take ? ov : bv;
            bi = take ? oi : bi;
        }
        bestI[r] = bi;
    }
    if (l15 == 0) {
#pragma unroll
        for (int r = 0; r < 8; ++r)
            negSh[wave * 16 + half * 8 + r] = bestI[r];
    }
    __syncthreads();

    // Distances + loss per row; A rows still resident in LDS.
    for (int rr = 0; rr < 16; ++rr) {
        const int localRow = wave * 16 + rr;
        const int gRow = rowBlock + localRow;
        const int lab  = label[gRow];
        const int ni   = negSh[localRow];
        const float* __restrict__ yrow = prototypes + (size_t)lab * DK;
        const float* __restrict__ nrow = ftr + (size_t)ni * DK;
        const float* frow = As + (size_t)localRow * LDA;
        float pd = 0.f, nd = 0.f;
#pragma unroll
        for (int j = 0; j < 8; ++j) {
            const int d   = lane * 8 + j;
            const float f  = frow[d];
            const float dy = f - yrow[d];
            const float dn = f - nrow[d];
            pd += dy * dy;
            nd += dn * dn;
        }
#pragma unroll
        for (int off = 16; off > 0; off >>= 1) {
            pd += __shfl_xor(pd, off, 32);
            nd += __shfl_xor(nd, off, 32);
        }
        if (lane == 0)
            losses[gRow] = fmaxf(pd - nd + 0.5f, 0.f);
    }
}

// Deterministic fixed-order reduction: mean over the 4096 per-row losses.
__global__ __launch_bounds__(256)
void SpaceLoss2_reduce_kernel(const float* __restrict__ losses,
                              float* __restrict__ out)
{
    __shared__ float s[256];
    const int t = threadIdx.x;
    float v = 0.f;
    for (int i = t; i < BN; i += 256) v += losses[i];
    s[t] = v;
    __syncthreads();
    for (int off = 128; off > 0; off >>= 1) {
        if (t < off) s[t] += s[t + off];
        __syncthreads();
    }
    if (t == 0) out[0] = s[0] * (1.0f / (float)BN);
}

extern "C" void kernel_launch(void* const* d_in, const int* in_sizes, int n_in,
                              void* d_out, int out_size, void* d_ws, size_t ws_size,
                              hipStream_t stream)
{
    (void)in_sizes; (void)n_in; (void)out_size; (void)ws_size;
    const float* ftr        = (const float*)d_in[0];
    // d_in[1] = teachor_ftr: unused by the reference computation.
    const float* prototypes = (const float*)d_in[2];
    const int*   label      = (const int*)d_in[3];
    float* losses = (float*)d_ws;                 // 4096 floats of scratch
    float* out    = (float*)d_out;

    const size_t shmem = SMEM_FLOATS * sizeof(float);
    SpaceLoss2_wmma_kernel<<<BN / ROWS_PB, NTHREADS, shmem, stream>>>(
        ftr, prototypes, label, losses);
    SpaceLoss2_reduce_kernel<<<1, 256, 0, stream>>>(losses, out);
}